// NormXCorr_41412074668366
// MI455X (gfx1250) — compile-verified
//
#include <hip/hip_runtime.h>
#include <math.h>

// normxcorr cost volume on gfx1250 using full-precision V_WMMA_F32_16X16X4_F32.
// X,Y: [64,25,37,12] f32 -> out: [64, 25*5*12, 37, 12] f32.
// Store-bandwidth-bound (170 MB out / 23.3 TB/s); fp32 WMMA keeps reference precision.

#define BATCH 64
#define CH    25
#define HH    37
#define WW    12
#define HW    (HH * WW)   // 444
#define PAT   25          // 5x5 patch
#define PP    28          // padded K (7 chunks of 4)
#define NPIX  (HH * WW)   // 444
#define NPIXP 448         // over-allocated rows so lane 12..15 reads stay in LDS
#define NSH   5

typedef float v2f __attribute__((ext_vector_type(2)));
typedef float v8f __attribute__((ext_vector_type(8)));

__global__ __launch_bounds__(256)
void normxcorr_wmma_f32(const float* __restrict__ X,
                        const float* __restrict__ Y,
                        float* __restrict__ out) {
  extern __shared__ float smem[];
  float* rawX = smem;                 // 448
  float* rawY = rawX + NPIXP;         // 448
  float* Xn   = rawY + NPIXP;         // 448*28  (pre-scaled by 1/25)
  float* Yn   = Xn + NPIXP * PP;      // 448*28

  const int bc = blockIdx.x;          // b*CH + c
  const int t  = threadIdx.x;
  const float* Xg = X + (size_t)bc * NPIX;
  const float* Yg = Y + (size_t)bc * NPIX;

  // Stage the raw (b,c) slices into LDS.
  for (int idx = t; idx < NPIX; idx += 256) {
    rawX[idx] = Xg[idx];
    rawY[idx] = Yg[idx];
  }
  __syncthreads();

  // Per-pixel patch extraction + normalization, written K-padded to 28.
  for (int pix = t; pix < NPIXP; pix += 256) {
    float* xo = Xn + pix * PP;
    float* yo = Yn + pix * PP;
    if (pix < NPIX) {
      const int i = pix / WW, j = pix % WW;
      float vx[PAT], vy[PAT];
      float sx = 0.f, sy = 0.f;
#pragma unroll
      for (int di = 0; di < 5; ++di) {
#pragma unroll
        for (int dj = 0; dj < 5; ++dj) {
          const int ii = i + di - 2, jj = j + dj - 2;
          const bool in = (ii >= 0) & (ii < HH) & (jj >= 0) & (jj < WW);
          const float fx = in ? rawX[ii * WW + jj] : 0.f;
          const float fy = in ? rawY[ii * WW + jj] : 0.f;
          vx[di * 5 + dj] = fx;  vy[di * 5 + dj] = fy;
          sx += fx;              sy += fy;
        }
      }
      const float mx = sx * (1.f / PAT), my = sy * (1.f / PAT);
      float qx = 0.f, qy = 0.f;
#pragma unroll
      for (int p = 0; p < PAT; ++p) {
        vx[p] -= mx;  vy[p] -= my;
        qx += vx[p] * vx[p];  qy += vy[p] * vy[p];
      }
      // fold the final 1/25 cost-volume scale into Xn only
      const float rx = (1.f / PAT) / (sqrtf(qx * (1.f / PAT)) + 0.01f);
      const float ry = 1.f        / (sqrtf(qy * (1.f / PAT)) + 0.01f);
#pragma unroll
      for (int p = 0; p < PAT; ++p) { xo[p] = vx[p] * rx;  yo[p] = vy[p] * ry; }
      xo[25] = xo[26] = xo[27] = 0.f;
      yo[25] = yo[26] = yo[27] = 0.f;
    } else {
#pragma unroll
      for (int p = 0; p < PP; ++p) { xo[p] = 0.f; yo[p] = 0.f; }
    }
  }
  __syncthreads();

  // GEMM phase: D[k][j] = sum_p Yn_d[k][p] * Xn[j][p], K=28 as 7 x (K=4) WMMAs.
  // A (16x4, M=k): lane<16 holds A[m][kb+0..1], lane>=16 holds A[m][kb+2..3].
  // B (4x16, N=j): lane<16 holds B[kb+0..1][n], lane>=16 holds B[kb+2..3][n].
  const int wave = t >> 5;
  const int lane = t & 31;
  const int ln   = lane & 15;   // m (=k) for A, n (=j) for B
  const int hi   = lane >> 4;
  const int ko   = hi * 2;

  for (int i = wave; i < HH; i += 8) {
    v2f bfrag[7];
    const float* xrow = Xn + (i * WW + ln) * PP + ko;
#pragma unroll
    for (int kk = 0; kk < 7; ++kk)
      bfrag[kk] = *(const v2f*)(xrow + kk * 4);

    for (int d = 0; d < NSH; ++d) {
      const int iy = i + d - 2;                    // vertical shift (wave-uniform)
      v8f acc = {0.f, 0.f, 0.f, 0.f, 0.f, 0.f, 0.f, 0.f};
      if (iy >= 0 && iy < HH) {                    // uniform branch: EXEC all-ones at WMMA
        const float* yrow = Yn + (iy * WW + ln) * PP + ko;
#pragma unroll
        for (int kk = 0; kk < 7; ++kk) {
          v2f afrag = *(const v2f*)(yrow + kk * 4);
          acc = __builtin_amdgcn_wmma_f32_16x16x4_f32(
              false, afrag, false, bfrag[kk], (short)0, acc, false, false);
        }
      }
      // D layout: VGPR r, lane<16 -> k=r, lane>=16 -> k=r+8; n=lane&15 = j.
      // k = r + 8*hi. r=0..3 always valid; r=4..7 valid only for hi==0.
      // Two uniform-predicate store blocks instead of per-r branches.
      const int j = ln;
      float* op = out +
          (((((size_t)bc * NSH + d) * WW + 8 * hi) * HH + i) * WW + j);
      if (j < WW) {
        __builtin_nontemporal_store(acc[0], op + 0 * HW);
        __builtin_nontemporal_store(acc[1], op + 1 * HW);
        __builtin_nontemporal_store(acc[2], op + 2 * HW);
        __builtin_nontemporal_store(acc[3], op + 3 * HW);
        if (hi == 0) {
          __builtin_nontemporal_store(acc[4], op + 4 * HW);
          __builtin_nontemporal_store(acc[5], op + 5 * HW);
          __builtin_nontemporal_store(acc[6], op + 6 * HW);
          __builtin_nontemporal_store(acc[7], op + 7 * HW);
        }
      }
    }
  }
}

extern "C" void kernel_launch(void* const* d_in, const int* in_sizes, int n_in,
                              void* d_out, int out_size, void* d_ws, size_t ws_size,
                              hipStream_t stream) {
  (void)in_sizes; (void)n_in; (void)out_size; (void)d_ws; (void)ws_size;
  const float* X = (const float*)d_in[0];
  const float* Y = (const float*)d_in[1];
  float* out = (float*)d_out;
  const size_t smem_bytes =
      (size_t)(2 * NPIXP + 2 * NPIXP * PP) * sizeof(float);  // 103,936 B
  normxcorr_wmma_f32<<<BATCH * CH, 256, smem_bytes, stream>>>(X, Y, out);
}